// Net_12867722019590
// MI455X (gfx1250) — compile-verified
//
#include <hip/hip_runtime.h>

typedef __attribute__((ext_vector_type(2)))  float v2f;
typedef __attribute__((ext_vector_type(8)))  float v8f;
typedef __attribute__((ext_vector_type(4)))  unsigned int v4u;
typedef __attribute__((ext_vector_type(8)))  int v8i;
typedef __attribute__((ext_vector_type(4)))  int v4i;

#define FIN 128   // input feature dim
#define FC  60    // conv output features
#define FCP 64    // padded conv features (256B rows)

// ---------------------------------------------------------------------------
// Tensor Data Mover: 1-D tile load (nElems4B fp32 words) global -> LDS.
// D# packed per CDNA5 ISA ch.8: group0 = {count, lds_addr, global_addr, type=2},
// group1 = {data_size=4B, tensor_dim0=tile_dim0=nElems, dims1=1, stride0=nElems}.
// Groups 2/3 (and the trailing group on the clang-23 6-arg builtin) are zero
// (<=2D tensor). Issued once per block (EXEC-ignored op), tracked by TENSORcnt.
// ---------------------------------------------------------------------------
__device__ __forceinline__ void tdm_load_1d(unsigned lds_off, const void* gptr,
                                            unsigned nElems4B) {
  unsigned long long ga = (unsigned long long)gptr;
  v4u g0;
  g0[0] = 1u;                                                  // count=1
  g0[1] = lds_off;                                             // lds_addr (bytes)
  g0[2] = (unsigned)(ga & 0xFFFFFFFFu);                        // global_addr lo
  g0[3] = (unsigned)((ga >> 32) & 0x01FFFFFFu) | 0x80000000u;  // addr hi | type=2
  v8i g1;
  g1[0] = 0x20000;                                   // data_size = 2 (4 bytes)
  g1[1] = (int)((nElems4B & 0xFFFFu) << 16);         // tensor_dim0 [15:0]
  g1[2] = (int)((nElems4B >> 16) | (1u << 16));      // tensor_dim0 hi | tensor_dim1=1
  g1[3] = (int)(nElems4B << 16);                     // tile_dim0 (<=65535)
  g1[4] = 1;                                         // tile_dim1=1, tile_dim2=0
  g1[5] = (int)nElems4B;                             // tensor_dim0_stride lo
  g1[6] = 0;
  g1[7] = 0;
  v4i z4 = {0, 0, 0, 0};
  v8i z8 = {0, 0, 0, 0, 0, 0, 0, 0};
  __builtin_amdgcn_tensor_load_to_lds(g0, g1, z4, z4, z8, 0);
}

// ---------------------------------------------------------------------------
__global__ void k_zero(float4* __restrict__ p, int tot4) {
  int g = blockIdx.x * blockDim.x + threadIdx.x;
  if (g < tot4) p[g] = make_float4(0.f, 0.f, 0.f, 0.f);
}

// Build transposed, padded fp32 conv weights with BN scale folded in:
//   WpT[n][k] = W_conv[k][n] * gamma[n]*rsqrt(var[n]+eps)      (n<60, else 0)
//   cB[n]     = (b_conv[n]-mean[n])*scale[n] + beta[n]         (n<60, else 0)
// so conv output is h[:,n] = (x @ Wp)[:,n] + cB[n], exactly matching BN inference.
__global__ void k_prep_w(const float* __restrict__ Wc, const float* __restrict__ bc,
                         const float* __restrict__ gm, const float* __restrict__ bt,
                         const float* __restrict__ mn, const float* __restrict__ vr,
                         float* __restrict__ WpT, float* __restrict__ cB) {
  int tid = threadIdx.x;
  for (int i = tid; i < FCP * FIN; i += blockDim.x) {
    int n = i >> 7, k = i & 127;
    float v = 0.f;
    if (n < FC) v = Wc[k * FC + n] * (gm[n] * rsqrtf(vr[n] + 1e-3f));
    WpT[i] = v;
  }
  if (tid < FCP) {
    if (tid < FC) {
      float s = gm[tid] * rsqrtf(vr[tid] + 1e-3f);
      cB[tid] = (bc[tid] - mn[tid]) * s + bt[tid];
    } else {
      cB[tid] = 0.f;   // padded features stay exactly zero
    }
  }
}

// ---------------------------------------------------------------------------
// Conv GEMM in exact fp32 via V_WMMA_F32_16X16X4_F32.
// Block = 128 threads = 4 waves; covers 16 nodes x 64 features; wave w owns
// feature tile [w*16, w*16+16). K = 128 = 32 steps of K=4.
// Operands staged in LDS by the Tensor Data Mover:
//   L[0..2047]    = x tile   [16 nodes][128]   (8 KB)
//   L[2048..]     = WpT      [64 feats][128]   (32 KB)
// A layout (32-bit A 16x4): lane m=lane&15 is row M; VGPR v holds K=2*hi+v.
// B layout mirrors A with N on lanes: lane n, VGPR v holds K=2*hi+v.
// C/D layout: VGPR r holds M=r+8*hi, N=lane&15.
// Requires numNodes % 16 == 0 (N = 100000 = 6250*16).
// ---------------------------------------------------------------------------
__global__ __launch_bounds__(128) void k_conv(const float* __restrict__ x,
                                              const float* __restrict__ WpT,
                                              const float* __restrict__ cB,
                                              float* __restrict__ h) {
  __shared__ float L[2048 + 8192];
  const int lane = threadIdx.x & 31;
  const int w    = threadIdx.x >> 5;
  const int m    = lane & 15;
  const int hi   = lane >> 4;
  const int node0 = blockIdx.x * 16;

  if (threadIdx.x == 0) {
    unsigned lds0 = (unsigned)(unsigned long long)&L[0];  // low 32 bits = LDS offset
    tdm_load_1d(lds0, x + (long)node0 * FIN, 2048);
    tdm_load_1d(lds0 + 2048 * 4, WpT, 8192);
    __builtin_amdgcn_s_wait_tensorcnt(0);
  }
  __syncthreads();

  const float* ar = &L[m * FIN];                    // x row for this lane's M
  const float* br = &L[2048 + (w * 16 + m) * FIN];  // W^T row for this lane's N
  v8f c = {};
#pragma unroll
  for (int kt = 0; kt < 32; ++kt) {
    v2f a = *(const v2f*)(ar + kt * 4 + 2 * hi);
    v2f b = *(const v2f*)(br + kt * 4 + 2 * hi);
    c = __builtin_amdgcn_wmma_f32_16x16x4_f32(false, a, false, b,
                                              (short)0, c, false, false);
  }

  const int f = w * 16 + m;
  const float sB = cB[f];
#pragma unroll
  for (int r = 0; r < 8; ++r) {
    int node = node0 + r + 8 * hi;
    h[(long)node * FCP + f] = c[r] + sB;
  }
}

// ---------------------------------------------------------------------------
// Edge gather + scatter-add: 16 lanes per edge, lane c owns float4 chunk c
// (c<15 covers the 60 real features). Coalesced b128 gather from h[src]
// (25.6 MB table -> L2 resident), global_atomic_add_f32 x4 into agg[dst].
// ---------------------------------------------------------------------------
__global__ __launch_bounds__(256) void k_scatter(const float* __restrict__ h,
                                                 const int* __restrict__ esrc,
                                                 const int* __restrict__ edst,
                                                 float* __restrict__ agg, int nE) {
  int gid = blockIdx.x * blockDim.x + threadIdx.x;
  int e = gid >> 4, c = gid & 15;
  if (e < nE && c < 15) {
    int s = esrc[e], d = edst[e];
    const float4 v = *(const float4*)(h + (long)s * FCP + c * 4);
    float* p = agg + (long)d * FCP + c * 4;
    atomicAdd(p + 0, v.x);
    atomicAdd(p + 1, v.y);
    atomicAdd(p + 2, v.z);
    atomicAdd(p + 3, v.w);
  }
}

// ---------------------------------------------------------------------------
// PReLU + dense chain, one node per thread, all weights staged in LDS (30 KB).
// Fully unrolled so activations stay in VGPRs.
// ---------------------------------------------------------------------------
template <int FI, int FO, bool SW>
__device__ __forceinline__ void layerf(const float* in, float* out,
                                       const float* Wl, const float* bl) {
#pragma unroll
  for (int j = 0; j < FO; ++j) {
    float acc = bl[j];
#pragma unroll
    for (int i = 0; i < FI; ++i) acc = fmaf(in[i], Wl[i * FO + j], acc);
    if (SW) acc = acc * (1.f / (1.f + __expf(-acc)));   // swish
    out[j] = acc;
  }
}

// LDS offsets (floats)
#define OD1W 0
#define OD1B 3000
#define OD2W 3050
#define OD2B 5050
#define OD3W 5090
#define OD3B 6290
#define OD31W 6320
#define OD31B 6920
#define OD32W 6940
#define OD32B 7140
#define ODVW 7150
#define ODVB 7450
#define OD4W 7480
#define OD4B 7630
#define OD41W 7635
#define OD41B 7645
#define OD5W 7647
#define OD5B 7649
#define OALPHA 7650
#define LDSTOT 7710

__global__ __launch_bounds__(128) void k_tail(
    const float* __restrict__ agg, const float* __restrict__ alpha,
    const float* W1, const float* b1, const float* W2, const float* b2,
    const float* W3, const float* b3, const float* W31, const float* b31,
    const float* W32, const float* b32, const float* Wv, const float* bv,
    const float* W4, const float* b4, const float* W41, const float* b41,
    const float* W5, const float* b5, float* __restrict__ out, int nN) {
  __shared__ float L[LDSTOT];
  const int tid = threadIdx.x, bs = blockDim.x;
  auto cp = [&](const float* src, int off, int n) {
    for (int i = tid; i < n; i += bs) L[off + i] = src[i];
  };
  cp(W1, OD1W, 3000);  cp(b1, OD1B, 50);
  cp(W2, OD2W, 2000);  cp(b2, OD2B, 40);
  cp(W3, OD3W, 1200);  cp(b3, OD3B, 30);
  cp(W31, OD31W, 600); cp(b31, OD31B, 20);
  cp(W32, OD32W, 200); cp(b32, OD32B, 10);
  cp(Wv, ODVW, 300);   cp(bv, ODVB, 30);
  cp(W4, OD4W, 150);   cp(b4, OD4B, 5);
  cp(W41, OD41W, 10);  cp(b41, OD41B, 2);
  cp(W5, OD5W, 2);     cp(b5, OD5B, 1);
  cp(alpha, OALPHA, 60);
  __syncthreads();

  int node = blockIdx.x * bs + tid;
  if (node >= nN) return;

  float A[60], B[50];
#pragma unroll
  for (int f = 0; f < 60; ++f) {
    float v = agg[(long)node * FCP + f];
    A[f] = (v >= 0.f) ? v : L[OALPHA + f] * v;   // PReLU
  }
  layerf<60, 50, true >(A, B, &L[OD1W],  &L[OD1B]);
  layerf<50, 40, false>(B, A, &L[OD2W],  &L[OD2B]);
  layerf<40, 30, true >(A, B, &L[OD3W],  &L[OD3B]);
  layerf<30, 20, false>(B, A, &L[OD31W], &L[OD31B]);
  layerf<20, 10, true >(A, B, &L[OD32W], &L[OD32B]);
  layerf<10, 30, false>(B, A, &L[ODVW],  &L[ODVB]);
  layerf<30, 5,  false>(A, B, &L[OD4W],  &L[OD4B]);
  layerf<5,  2,  false>(B, A, &L[OD41W], &L[OD41B]);
  layerf<2,  1,  false>(A, B, &L[OD5W],  &L[OD5B]);

  out[node] = rintf(B[0] * 1000.f) / 1000.f;   // round half-to-even, 3 decimals
}

// ---------------------------------------------------------------------------
extern "C" void kernel_launch(void* const* d_in, const int* in_sizes, int n_in,
                              void* d_out, int out_size, void* d_ws, size_t ws_size,
                              hipStream_t stream) {
  const float* x    = (const float*)d_in[0];
  const int*   esrc = (const int*)d_in[1];
  const int*   edst = (const int*)d_in[2];
  const float* Wc   = (const float*)d_in[3];
  const float* bc   = (const float*)d_in[4];
  const float* W1 = (const float*)d_in[5],  *b1 = (const float*)d_in[6];
  const float* W2 = (const float*)d_in[7],  *b2 = (const float*)d_in[8];
  const float* W3 = (const float*)d_in[9],  *b3 = (const float*)d_in[10];
  const float* W31 = (const float*)d_in[11], *b31 = (const float*)d_in[12];
  const float* W32 = (const float*)d_in[13], *b32 = (const float*)d_in[14];
  const float* Wv = (const float*)d_in[15], *bv = (const float*)d_in[16];
  const float* W4 = (const float*)d_in[17], *b4 = (const float*)d_in[18];
  const float* W41 = (const float*)d_in[19], *b41 = (const float*)d_in[20];
  const float* W5 = (const float*)d_in[21], *b5 = (const float*)d_in[22];
  const float* gm = (const float*)d_in[23];
  const float* bt = (const float*)d_in[24];
  const float* mn = (const float*)d_in[25];
  const float* vr = (const float*)d_in[26];
  const float* al = (const float*)d_in[27];

  const int nN = in_sizes[0] / FIN;   // 100000 (divisible by 16)
  const int nE = in_sizes[1];         // 3200000

  // Workspace layout (all offsets 256B-aligned): ~51.3 MB total
  char* ws = (char*)d_ws;
  size_t oWpT = 0;                                        // fp32 W^T*scale  32 KB
  size_t oCB  = oWpT + (size_t)FCP * FIN * sizeof(float); // 32768
  size_t oH   = oCB + 256;                                // h[N][64]    25.6 MB
  size_t oAgg = oH + (size_t)nN * FCP * sizeof(float);    // agg[N][64]  25.6 MB
  float* WpT = (float*)(ws + oWpT);
  float* cB  = (float*)(ws + oCB);
  float* h   = (float*)(ws + oH);
  float* agg = (float*)(ws + oAgg);

  k_prep_w<<<1, 256, 0, stream>>>(Wc, bc, gm, bt, mn, vr, WpT, cB);
  int t4a = nN * FCP / 4;
  k_zero<<<(t4a + 255) / 256, 256, 0, stream>>>((float4*)agg, t4a);

  k_conv<<<nN / 16, 128, 0, stream>>>(x, WpT, cB, h);

  long sThreads = (long)nE * 16;
  k_scatter<<<(int)((sThreads + 255) / 256), 256, 0, stream>>>(h, esrc, edst, agg, nE);

  k_tail<<<(nN + 127) / 128, 128, 0, stream>>>(
      agg, al, W1, b1, W2, b2, W3, b3, W31, b31, W32, b32, Wv, bv, W4, b4,
      W41, b41, W5, b5, (float*)d_out, nN);
}